// QuantumNeuralNetwork_69526930588460
// MI455X (gfx1250) — compile-verified
//
#include <hip/hip_runtime.h>
#include <hip/hip_bf16.h>

// ---------------------------------------------------------------------------
// QNN state-vector simulator for gfx1250 (MI455X), FP32 WMMA formulation.
//
// State (4096 reals) lives in LDS; each RY layer = 3 mode contractions with
// 16x16 real matrices executed as chained V_WMMA_F32_16X16X4_F32.
// CNOT-ring permutation is a fixed gather folded into pass-1 loads.
// ---------------------------------------------------------------------------

#define NQ      12
#define DIM     4096
#define NLAYERS 16
#define BATCH   2048
#define NCLASS  5

typedef float v2f __attribute__((ext_vector_type(2)));
typedef float v8f __attribute__((ext_vector_type(8)));

// LDS padding: one extra dword per 64 to de-conflict stride-16 access.
__device__ __forceinline__ int pa(int i) { return i + (i >> 6); }
#define PADDED_DIM (DIM + (DIM >> 6))   // 4160

// ---------------------------------------------------------------------------
// Kernel 1: build per-layer 16x16 gate matrices  A = R(4g)⊗R(4g+1)⊗R(4g+2)⊗R(4g+3)
// gmat layout: [layer][group(3)][m*16+k], 48 * 256 floats.
// ---------------------------------------------------------------------------
__global__ __launch_bounds__(256) void qnn_gates(const float* __restrict__ thetas,
                                                 float* __restrict__ gmat) {
  const int blk = blockIdx.x;          // 0..47  = l*3 + g
  const int l = blk / 3, g = blk % 3;
  const int t = threadIdx.x;           // 0..255 = m*16 + k
  const int m = t >> 4, k = t & 15;
  float prod = 1.0f;
#pragma unroll
  for (int j = 0; j < 4; ++j) {
    const int q = g * 4 + j;
    const float th = 0.5f * thetas[l * NQ + q];
    const float c = cosf(th), s = sinf(th);
    const int mb = (m >> (3 - j)) & 1;
    const int kb = (k >> (3 - j)) & 1;
    // RY = [[c, -s], [s, c]]
    const float f = mb ? (kb ? c : s) : (kb ? -s : c);
    prod *= f;
  }
  gmat[blk * 256 + t] = prod;
}

// ---------------------------------------------------------------------------
// Kernel 2: main circuit. One 128-thread block (4 waves) per real vector.
// blockIdx.x in [0, 2*BATCH): row = batch*2 + component(0=re,1=im).
// Writes 5 signed |psi|^2 sums per vector into part[row*5 + w].
// ---------------------------------------------------------------------------
__global__ __launch_bounds__(128) void qnn_main(const float* __restrict__ zr,
                                                const float* __restrict__ zi,
                                                const float* __restrict__ gmat,
                                                float* __restrict__ part) {
  __shared__ float bufA[PADDED_DIM];
  __shared__ float bufB[PADDED_DIM];
  __shared__ unsigned short tbl[DIM];   // composed CNOT-ring gather table
  __shared__ float gm[3 * 256];         // this layer's three 16x16 matrices
  __shared__ float red[128 * NCLASS];   // reduction scratch

  const int tid  = threadIdx.x;
  const int row  = blockIdx.x;
  const int comp = row & 1;
  const int bidx = row >> 1;
  const float* __restrict__ src = comp ? zi : zr;

  // --- initial load: coalesced global -> padded LDS ---
  for (int t = tid; t < DIM; t += 128)
    bufA[pa(t)] = src[(size_t)bidx * DIM + t];

  // --- composed ring permutation: final[i] = prev[ perm_0(perm_1(...perm_11(i))) ]
  for (int i = tid; i < DIM; i += 128) {
    int j = i;
#pragma unroll
    for (int q = NQ - 1; q >= 0; --q) {
      const int cb = 1 << (NQ - 1 - q);
      const int tb = 1 << (NQ - 1 - ((q + 1) % NQ));
      j = (j & cb) ? (j ^ tb) : j;
    }
    tbl[i] = (unsigned short)j;
  }
  __syncthreads();

  float* cur = bufA;   // layer input
  float* alt = bufB;   // scratch / layer output

  const int lane = tid & 31;
  const int wv   = tid >> 5;      // wave 0..3
  const int half = lane >> 4;     // 0: lanes 0-15, 1: lanes 16-31
  const int hm   = lane & 15;     // row (A) / column (B,C,D) index

  for (int l = 0; l < NLAYERS; ++l) {
    // stage this layer's gate matrices
    for (int t = tid; t < 3 * 256; t += 128) gm[t] = gmat[l * 768 + t];
    __syncthreads();

    // ================= pass 1: contract mode a (stride 256) ==============
    {
      const float* in = cur;
      float* out = alt;
      v2f a[4];
#pragma unroll
      for (int kk = 0; kk < 4; ++kk) {
        const int k0 = kk * 4 + half * 2;
        a[kk].x = gm[hm * 16 + k0];
        a[kk].y = gm[hm * 16 + k0 + 1];
      }
#pragma unroll
      for (int t4 = 0; t4 < 4; ++t4) {
        const int bcb = (wv * 4 + t4) * 16;     // column-tile base over (b,c)
        v8f acc = {0.f, 0.f, 0.f, 0.f, 0.f, 0.f, 0.f, 0.f};
#pragma unroll
        for (int kk = 0; kk < 4; ++kk) {
          const int k0 = kk * 4 + half * 2;
          int i0 = k0 * 256 + bcb + hm;
          int i1 = i0 + 256;
          if (l > 0) { i0 = tbl[i0]; i1 = tbl[i1]; }   // fold CNOT ring
          v2f bv;
          bv.x = in[pa(i0)];
          bv.y = in[pa(i1)];
          acc = __builtin_amdgcn_wmma_f32_16x16x4_f32(
              false, a[kk], false, bv, (short)0, acc, false, false);
        }
#pragma unroll
        for (int r = 0; r < 8; ++r)
          out[pa((r + half * 8) * 256 + bcb + hm)] = acc[r];
      }
    }
    __syncthreads();

    // ================= pass 2: contract mode b (stride 16) ===============
    {
      const float* in = alt;
      float* out = cur;
      v2f a[4];
#pragma unroll
      for (int kk = 0; kk < 4; ++kk) {
        const int k0 = kk * 4 + half * 2;
        a[kk].x = gm[256 + hm * 16 + k0];
        a[kk].y = gm[256 + hm * 16 + k0 + 1];
      }
#pragma unroll
      for (int t4 = 0; t4 < 4; ++t4) {
        const int base = (wv * 4 + t4) * 256;   // fixed a; columns n = c
        v8f acc = {0.f, 0.f, 0.f, 0.f, 0.f, 0.f, 0.f, 0.f};
#pragma unroll
        for (int kk = 0; kk < 4; ++kk) {
          const int k0 = kk * 4 + half * 2;
          const int i0 = base + k0 * 16 + hm;
          v2f bv;
          bv.x = in[pa(i0)];
          bv.y = in[pa(i0 + 16)];
          acc = __builtin_amdgcn_wmma_f32_16x16x4_f32(
              false, a[kk], false, bv, (short)0, acc, false, false);
        }
#pragma unroll
        for (int r = 0; r < 8; ++r)
          out[pa(base + (r + half * 8) * 16 + hm)] = acc[r];
      }
    }
    __syncthreads();

    // ================= pass 3: contract mode c (stride 1) ================
    {
      const float* in = cur;
      float* out = alt;
      v2f a[4];
#pragma unroll
      for (int kk = 0; kk < 4; ++kk) {
        const int k0 = kk * 4 + half * 2;
        a[kk].x = gm[512 + hm * 16 + k0];
        a[kk].y = gm[512 + hm * 16 + k0 + 1];
      }
#pragma unroll
      for (int t4 = 0; t4 < 4; ++t4) {
        const int base = (wv * 4 + t4) * 256;   // fixed a; columns n = b
        v8f acc = {0.f, 0.f, 0.f, 0.f, 0.f, 0.f, 0.f, 0.f};
#pragma unroll
        for (int kk = 0; kk < 4; ++kk) {
          const int k0 = kk * 4 + half * 2;
          const int i0 = base + hm * 16 + k0;
          v2f bv;
          bv.x = in[pa(i0)];
          bv.y = in[pa(i0 + 1)];
          acc = __builtin_amdgcn_wmma_f32_16x16x4_f32(
              false, a[kk], false, bv, (short)0, acc, false, false);
        }
#pragma unroll
        for (int r = 0; r < 8; ++r)
          out[pa(base + hm * 16 + r + half * 8)] = acc[r];
      }
    }
    // layer output now in `alt`; swap for next layer
    {
      float* tmp = cur; cur = alt; alt = tmp;
    }
    __syncthreads();
  }

  // --- signed |psi|^2 reduction (final ring permutation folded via tbl) ---
  float acc[NCLASS] = {0.f, 0.f, 0.f, 0.f, 0.f};
  const float* fin = cur;
  for (int i = tid; i < DIM; i += 128) {
    const float s = fin[pa(tbl[i])];
    const float v = s * s;
#pragma unroll
    for (int w = 0; w < NCLASS; ++w)
      acc[w] += ((i >> (NQ - 1 - w)) & 1) ? -v : v;
  }
#pragma unroll
  for (int w = 0; w < NCLASS; ++w) red[tid * NCLASS + w] = acc[w];
  __syncthreads();
  for (int s = 64; s > 0; s >>= 1) {
    if (tid < s) {
#pragma unroll
      for (int w = 0; w < NCLASS; ++w)
        red[tid * NCLASS + w] += red[(tid + s) * NCLASS + w];
    }
    __syncthreads();
  }
  if (tid < NCLASS) part[(size_t)row * NCLASS + tid] = red[tid];
}

// ---------------------------------------------------------------------------
// Kernel 3: logits = re-part + im-part, log-softmax NLL, mean.
// d_out[0] = loss, d_out[1 + b*5 + w] = logits.
// ---------------------------------------------------------------------------
__global__ __launch_bounds__(256) void qnn_loss(const float* __restrict__ part,
                                                const int* __restrict__ y,
                                                float* __restrict__ out) {
  __shared__ float sred[256];
  float lsum = 0.0f;
  for (int b = threadIdx.x; b < BATCH; b += 256) {
    float lg[NCLASS];
#pragma unroll
    for (int w = 0; w < NCLASS; ++w) {
      lg[w] = part[(size_t)(2 * b) * NCLASS + w] +
              part[(size_t)(2 * b + 1) * NCLASS + w];
      out[1 + b * NCLASS + w] = lg[w];
    }
    float mx = lg[0];
#pragma unroll
    for (int w = 1; w < NCLASS; ++w) mx = fmaxf(mx, lg[w]);
    float se = 0.0f;
#pragma unroll
    for (int w = 0; w < NCLASS; ++w) se += expf(lg[w] - mx);
    const float lse = mx + logf(se);
    const int t = y[b];
    lsum += lse - lg[t];
  }
  sred[threadIdx.x] = lsum;
  __syncthreads();
  for (int s = 128; s > 0; s >>= 1) {
    if (threadIdx.x < s) sred[threadIdx.x] += sred[threadIdx.x + s];
    __syncthreads();
  }
  if (threadIdx.x == 0) out[0] = sred[0] / (float)BATCH;
}

// ---------------------------------------------------------------------------
extern "C" void kernel_launch(void* const* d_in, const int* in_sizes, int n_in,
                              void* d_out, int out_size, void* d_ws, size_t ws_size,
                              hipStream_t stream) {
  (void)in_sizes; (void)n_in; (void)out_size; (void)ws_size;
  const float* zr = (const float*)d_in[0];     // (2048, 4096) f32
  const float* zi = (const float*)d_in[1];     // (2048, 4096) f32
  const float* th = (const float*)d_in[2];     // (16, 12)     f32
  const int*   y  = (const int*)d_in[3];       // (2048,)      int
  float* out = (float*)d_out;                  // 1 + 2048*5 f32

  float* gmat = (float*)d_ws;                  // 48*256 floats
  float* part = gmat + 48 * 256;               // 4096*5 floats

  hipLaunchKernelGGL(qnn_gates, dim3(48), dim3(256), 0, stream, th, gmat);
  hipLaunchKernelGGL(qnn_main, dim3(2 * BATCH), dim3(128), 0, stream,
                     zr, zi, gmat, part);
  hipLaunchKernelGGL(qnn_loss, dim3(1), dim3(256), 0, stream, part, y, out);
}